// CaT_36893769073058
// MI455X (gfx1250) — compile-verified
//
#include <hip/hip_runtime.h>
#include <math.h>

// Problem dims (fixed by the reference)
#define BB  16
#define NN  512
#define DD  256
#define LL  4
#define HH  8
#define HS  64
#define FFD 1024
#define TT  (BB * NN)   // 8192 tokens
#define HHS (HH * HS)   // 512

typedef float v2f __attribute__((ext_vector_type(2)));
typedef float v8f __attribute__((ext_vector_type(8)));

__device__ __forceinline__ v8f wmma4(v2f a, v2f b, v8f c) {
  // D = A(16x4,f32) x B(4x16,f32) + C(16x16,f32), full fp32 matrix pipe
  return __builtin_amdgcn_wmma_f32_16x16x4_f32(
      /*neg_a=*/false, a, /*neg_b=*/false, b,
      /*c_mod=*/(short)0, c, /*reuse_a=*/false, /*reuse_b=*/false);
}

__device__ __forceinline__ float swishf(float x) {
  return x / (1.0f + expf(-x));
}

// ---------------------------------------------------------------------------
// Build the two transposed DAG masks as fp32:
//   DT0[i,j] = clip(dag[j,i],0,1)            (layer 0)
//   DT1[i,j] = clip(dag[j,i] + (i==j),0,1)   (layers 1..3)
// ---------------------------------------------------------------------------
__global__ void build_masks_kernel(const int* __restrict__ dag,
                                   float* __restrict__ DT0,
                                   float* __restrict__ DT1) {
  int idx = blockIdx.x * blockDim.x + threadIdx.x;
  if (idx >= NN * NN) return;
  int i = idx / NN, j = idx % NN;
  float d = fminf(fmaxf((float)dag[j * NN + i], 0.0f), 1.0f);
  DT0[idx] = d;
  DT1[idx] = fminf(d + ((i == j) ? 1.0f : 0.0f), 1.0f);
}

// ---------------------------------------------------------------------------
// Generic batched fp32 WMMA GEMM:  C = epi(A @ B + bias [+ addsrc])
// A: MxK row-major (lda), B: KxN row-major (ldb), C: MxN row-major (ldc).
// Block = 256 threads = 8 waves; block tile 128x64; waves in a 4x2 grid;
// each wave register-blocks a 32x32 tile = 2x2 WMMA fragments (4 v8f accums),
// so 8 LDS scalar reads feed 4 v_wmma issues.
// act: 0 = identity, 1 = swish.
// ---------------------------------------------------------------------------
__global__ __launch_bounds__(256)
void gemm_wmma_kernel(const float* __restrict__ A,  int lda,  long strideA,
                      const float* __restrict__ Bm, int ldb,  long strideB,
                      const float* __restrict__ bias,         long strideBias,
                      const float* __restrict__ addsrc, int ldadd, long strideAdd,
                      float* __restrict__ C,      int ldc,  long strideC,
                      int K, int act)
{
  __shared__ float sA[128][20];  // stride 20: float4-store aligned, 16-row
  __shared__ float sB[16][68];   // fragment reads hit distinct banks

  const int tid  = threadIdx.x;
  const int wave = tid >> 5;
  const int lane = tid & 31;
  const int lh   = lane >> 4;    // half-wave selector (k-pair)
  const int ll   = lane & 15;

  const int z = blockIdx.z;
  const float* Ab = A  + (size_t)z * (size_t)strideA;
  const float* Bb = Bm + (size_t)z * (size_t)strideB;
  float*       Cb = C  + (size_t)z * (size_t)strideC;

  const int m0   = blockIdx.x * 128;
  const int n0   = blockIdx.y * 64;
  const int rowt = wave >> 1;    // 0..3 : which 32-row slab
  const int colt = wave & 1;     // 0..1 : which 32-col slab
  const int rbas = rowt * 32;
  const int cbas = colt * 32;

  // cooperative-load indices (float4 granularity)
  const int ar0 = tid >> 2;             // rows 0..63   (q = tid)
  const int ar1 = (tid + 256) >> 2;     // rows 64..127 (q = tid+256)
  const int ak4 = (tid & 3) * 4;        // k offset 0/4/8/12
  const int bk  = tid >> 4;             // 0..15 rows of B tile
  const int bn4 = (tid & 15) * 4;       // 0..60 cols

  v8f acc[2][2];
#pragma unroll
  for (int i = 0; i < 2; ++i)
#pragma unroll
    for (int j = 0; j < 2; ++j)
      acc[i][j] = (v8f){0.f, 0.f, 0.f, 0.f, 0.f, 0.f, 0.f, 0.f};

  for (int kb = 0; kb < K; kb += 16) {
    // Stage A 128x16 and B 16x64 tiles into LDS.
    const float4 av0 = *(const float4*)(Ab + (size_t)(m0 + ar0) * lda + kb + ak4);
    const float4 av1 = *(const float4*)(Ab + (size_t)(m0 + ar1) * lda + kb + ak4);
    const float4 bv  = *(const float4*)(Bb + (size_t)(kb + bk) * ldb + n0 + bn4);
    *(float4*)&sA[ar0][ak4] = av0;
    *(float4*)&sA[ar1][ak4] = av1;
    *(float4*)&sB[bk][bn4]  = bv;
    if (kb + 16 < K) {  // prefetch next tiles -> global_prefetch_b8
      __builtin_prefetch(Ab + (size_t)(m0 + ar0) * lda + kb + 16 + ak4, 0, 3);
      __builtin_prefetch(Ab + (size_t)(m0 + ar1) * lda + kb + 16 + ak4, 0, 3);
      __builtin_prefetch(Bb + (size_t)(kb + 16 + bk) * ldb + n0 + bn4, 0, 3);
    }
    __syncthreads();

#pragma unroll
    for (int ks = 0; ks < 4; ++ks) {
      const int kk = ks * 4 + lh * 2;
      v2f a0, a1, b0, b1;
      a0.x = sA[rbas + ll][kk];
      a0.y = sA[rbas + ll][kk + 1];
      a1.x = sA[rbas + 16 + ll][kk];
      a1.y = sA[rbas + 16 + ll][kk + 1];
      b0.x = sB[kk][cbas + ll];
      b0.y = sB[kk + 1][cbas + ll];
      b1.x = sB[kk][cbas + 16 + ll];
      b1.y = sB[kk + 1][cbas + 16 + ll];
      acc[0][0] = wmma4(a0, b0, acc[0][0]);
      acc[0][1] = wmma4(a0, b1, acc[0][1]);
      acc[1][0] = wmma4(a1, b0, acc[1][0]);
      acc[1][1] = wmma4(a1, b1, acc[1][1]);
    }
    __syncthreads();
  }

  // Epilogue: C/D layout => row = r + 8*lh inside a 16x16 tile, col = ll
#pragma unroll
  for (int rt = 0; rt < 2; ++rt) {
#pragma unroll
    for (int ct = 0; ct < 2; ++ct) {
      const int crow = m0 + rbas + rt * 16 + 8 * lh;
      const int ccol = n0 + cbas + ct * 16 + ll;
      const float bval = bias ? bias[(size_t)z * strideBias + ccol] : 0.0f;
#pragma unroll
      for (int r = 0; r < 8; ++r) {
        const int row = crow + r;
        float v = acc[rt][ct][r] + bval;
        if (addsrc) v += addsrc[(size_t)z * strideAdd + (size_t)row * ldadd + ccol];
        if (act) v = swishf(v);
        Cb[(size_t)row * ldc + ccol] = v;
      }
    }
  }
}

// ---------------------------------------------------------------------------
// Attention core per (b, h, 16 query rows):
//   S  = Qp_rows @ K^T * rsqrt(hs)           (WMMA, into LDS)
//   Sp = DT * S; masked = (Sp==0 ? -inf : Sp); masked softmax + dead rows
//   Prow = P + DT_row                        (fuses P@V + DT@V)
//   O_rows = Prow @ V                        (WMMA, split-K over all 8 waves)
// Qp/K/V laid out (H, B, N, HS); O laid out (B*N, H*HS) for head concat.
// ---------------------------------------------------------------------------
__global__ __launch_bounds__(256)
void attn_wmma_kernel(const float* __restrict__ Qp,
                      const float* __restrict__ Kq,
                      const float* __restrict__ V,
                      const float* __restrict__ DT,
                      float* __restrict__ O)
{
  __shared__ float sS[16][NN + 8];
  __shared__ float sRed[16][68];   // split-K partial sums

  const int tid  = threadIdx.x;
  const int wave = tid >> 5;
  const int lane = tid & 31;
  const int lh   = lane >> 4;
  const int ll   = lane & 15;

  const int i0 = blockIdx.x * 16;
  const int h  = blockIdx.y;
  const int b  = blockIdx.z;

  const size_t hb = (size_t)(h * BB + b) * NN * HS;
  const float* Qph = Qp + hb;
  const float* Kh  = Kq + hb;
  const float* Vh  = V  + hb;

  // ---- Phase 1: raw scaled scores into LDS -------------------------------
  for (int jt = wave; jt < NN / 16; jt += 8) {
    v8f acc = {0.f, 0.f, 0.f, 0.f, 0.f, 0.f, 0.f, 0.f};
    const int j = jt * 16 + ll;
#pragma unroll
    for (int kb = 0; kb < HS; kb += 4) {
      const int kk = kb + 2 * lh;
      v2f a, bf;
      a.x = Qph[(size_t)(i0 + ll) * HS + kk];
      a.y = Qph[(size_t)(i0 + ll) * HS + kk + 1];
      const float2 kv = *(const float2*)(Kh + (size_t)j * HS + kk);
      bf.x = kv.x;
      bf.y = kv.y;
      acc = wmma4(a, bf, acc);
    }
#pragma unroll
    for (int r = 0; r < 8; ++r)
      sS[r + 8 * lh][jt * 16 + ll] = acc[r] * 0.125f;  // 1/sqrt(64)
  }
  __syncthreads();

  // ---- Phase 2: mask + softmax + add DT row (wave handles rows w, w+8) ---
  const float NEGINF = -__builtin_inff();
#pragma unroll
  for (int rr = 0; rr < 2; ++rr) {
    const int r  = wave + rr * 8;
    const int gi = i0 + r;
    float mv[16];
    float mx = NEGINF;
#pragma unroll
    for (int e = 0; e < 16; ++e) {
      const int j = lane + 32 * e;
      const float sp = DT[(size_t)gi * NN + j] * sS[r][j];
      const float m  = (sp == 0.0f) ? NEGINF : sp;
      mv[e] = m;
      mx = fmaxf(mx, m);
    }
    for (int off = 16; off > 0; off >>= 1)
      mx = fmaxf(mx, __shfl_xor(mx, off, 32));
    const bool dead = (mx == NEGINF);
    float pv[16];
    float sum = 0.0f;
#pragma unroll
    for (int e = 0; e < 16; ++e) {
      const float p = dead ? 0.0f : expf(mv[e] - mx);
      pv[e] = p;
      sum += p;
    }
    for (int off = 16; off > 0; off >>= 1)
      sum += __shfl_xor(sum, off, 32);
    const float inv = dead ? 0.0f : 1.0f / sum;
#pragma unroll
    for (int e = 0; e < 16; ++e) {
      const int j = lane + 32 * e;
      sS[r][j] = pv[e] * inv + DT[(size_t)gi * NN + j];  // P + DT  (O=(P+DT)@V)
    }
  }
  __syncthreads();

  // ---- Phase 3: O_rows(16x64) = sS(16x512) @ V(512x64), split-K ----------
  // Waves 0..3 take k in [0,256); waves 4..7 take k in [256,512) on the same
  // 16-column slabs; partials reduced through LDS. All 8 waves issue WMMA.
  {
    v8f acc = {0.f, 0.f, 0.f, 0.f, 0.f, 0.f, 0.f, 0.f};
    const int nc  = (wave & 3) * 16 + ll;
    const int kst = (wave >> 2) * (NN / 2);
    for (int kb = kst; kb < kst + NN / 2; kb += 4) {
      const int kk = kb + 2 * lh;
      v2f a, bf;
      a.x = sS[ll][kk];
      a.y = sS[ll][kk + 1];
      bf.x = Vh[(size_t)kk * HS + nc];
      bf.y = Vh[(size_t)(kk + 1) * HS + nc];
      acc = wmma4(a, bf, acc);
    }
    if (wave >= 4) {
#pragma unroll
      for (int r = 0; r < 8; ++r)
        sRed[r + 8 * lh][nc] = acc[r];
    }
    __syncthreads();
    if (wave < 4) {
#pragma unroll
      for (int r = 0; r < 8; ++r) {
        const int row = i0 + r + 8 * lh;
        O[(size_t)(b * NN + row) * HHS + h * HS + nc] =
            acc[r] + sRed[r + 8 * lh][nc];
      }
    }
  }
}

// ---------------------------------------------------------------------------
// Host-side orchestration
// ---------------------------------------------------------------------------
static inline void launch_gemm(hipStream_t s,
                               const float* A, int lda, long strideA,
                               const float* B, int ldb, long strideB,
                               const float* bias, long strideBias,
                               const float* add, int ldadd, long strideAdd,
                               float* C, int ldc, long strideC,
                               int M, int N, int K, int act, int batch) {
  dim3 grid(M / 128, N / 64, batch), blk(256);
  gemm_wmma_kernel<<<grid, blk, 0, s>>>(A, lda, strideA, B, ldb, strideB,
                                        bias, strideBias, add, ldadd, strideAdd,
                                        C, ldc, strideC, K, act);
}

extern "C" void kernel_launch(void* const* d_in, const int* in_sizes, int n_in,
                              void* d_out, int out_size, void* d_ws, size_t ws_size,
                              hipStream_t stream) {
  (void)in_sizes; (void)n_in; (void)out_size; (void)ws_size;

  const float* X   = (const float*)d_in[0];
  const int*   dag = (const int*)  d_in[1];
  const float* Wk  = (const float*)d_in[2];
  const float* bk  = (const float*)d_in[3];
  const float* Wq  = (const float*)d_in[4];
  const float* bq  = (const float*)d_in[5];
  const float* Wv  = (const float*)d_in[6];
  const float* bv  = (const float*)d_in[7];
  const float* Wp  = (const float*)d_in[8];
  const float* bp  = (const float*)d_in[9];
  const float* W1  = (const float*)d_in[10];
  const float* b1  = (const float*)d_in[11];
  const float* W2  = (const float*)d_in[12];
  const float* b2  = (const float*)d_in[13];
  const float* Wlm = (const float*)d_in[14];
  const float* blm = (const float*)d_in[15];

  // Workspace carve-up (floats). Total ~137 MB.
  float* ws = (float*)d_ws;
  float* DT0  = ws; ws += (size_t)NN * NN;
  float* DT1  = ws; ws += (size_t)NN * NN;
  float* Qb   = ws; ws += (size_t)HH * TT * HS;   // (H,B,N,HS)
  float* Kb   = ws; ws += (size_t)HH * TT * HS;
  float* Vb   = ws; ws += (size_t)HH * TT * HS;
  float* Qp   = ws; ws += (size_t)HH * TT * HS;
  float* Ob   = ws; ws += (size_t)TT * HHS;       // (B*N, H*HS)
  float* Xmha = ws; ws += (size_t)TT * DD;
  float* Ffb  = ws; ws += (size_t)TT * FFD;
  float* Xcur = ws; ws += (size_t)TT * DD;

  build_masks_kernel<<<(NN * NN + 255) / 256, 256, 0, stream>>>(dag, DT0, DT1);

  for (int l = 0; l < LL; ++l) {
    const float* Xs = (l == 0) ? X : Xcur;
    const float* DT = (l == 0) ? DT0 : DT1;
    const size_t wOff = (size_t)l * HH * DD * HS;
    const size_t bOff = (size_t)l * HH * HS;

    // K/Q/V: per-head GEMMs (batch over H), swish epilogue.
    launch_gemm(stream, Xs, DD, 0, Wk + wOff, HS, (long)DD * HS,
                bk + bOff, HS, nullptr, 0, 0,
                Kb, HS, (long)TT * HS, TT, HS, DD, /*act=*/1, HH);
    launch_gemm(stream, Xs, DD, 0, Wq + wOff, HS, (long)DD * HS,
                bq + bOff, HS, nullptr, 0, 0,
                Qb, HS, (long)TT * HS, TT, HS, DD, 1, HH);
    launch_gemm(stream, Xs, DD, 0, Wv + wOff, HS, (long)DD * HS,
                bv + bOff, HS, nullptr, 0, 0,
                Vb, HS, (long)TT * HS, TT, HS, DD, 1, HH);

    // Q' = DT @ Q  per (h,b): the reassociated dag^T@S contraction.
    launch_gemm(stream, DT, NN, 0, Qb, HS, (long)NN * HS,
                nullptr, 0, nullptr, 0, 0,
                Qp, HS, (long)NN * HS, NN, HS, NN, 0, HH * BB);

    // Masked-softmax attention + (P+DT)@V, writes head-concat layout.
    attn_wmma_kernel<<<dim3(NN / 16, HH, BB), 256, 0, stream>>>(Qp, Kb, Vb, DT, Ob);

    // Projection + swish.
    launch_gemm(stream, Ob, HHS, 0, Wp + (size_t)l * HHS * DD, DD, 0,
                bp + (size_t)l * DD, 0, nullptr, 0, 0,
                Xmha, DD, 0, TT, DD, HHS, 1, 1);

    // FF1 + swish.
    launch_gemm(stream, Xmha, DD, 0, W1 + (size_t)l * DD * FFD, FFD, 0,
                b1 + (size_t)l * FFD, 0, nullptr, 0, 0,
                Ffb, FFD, 0, TT, FFD, DD, 1, 1);

    // FF2 + bias + residual(mha).
    launch_gemm(stream, Ffb, FFD, 0, W2 + (size_t)l * FFD * DD, DD, 0,
                b2 + (size_t)l * DD, 0, Xmha, DD, 0,
                Xcur, DD, 0, TT, DD, FFD, 0, 1);
  }

  // lm head: out = Xcur @ Wlm + blm
  launch_gemm(stream, Xcur, DD, 0, Wlm, DD, 0, blm, 0, nullptr, 0, 0,
              (float*)d_out, DD, 0, TT, DD, DD, 0, 1);
}